// JSSPDynamicEmbedding_72765335929536
// MI455X (gfx1250) — compile-verified
//
#include <hip/hip_runtime.h>

// Problem constants (from reference): bs=128, M=64, O=4096, J=64, E=256
#define BS   128
#define MM   64
#define OO   4096
#define JJ   64
#define EE   256
#define INV_SCALE 0.001f

typedef __attribute__((ext_vector_type(2))) float v2f;
typedef __attribute__((ext_vector_type(8))) float v8f;

// One block per batch b. 256 threads = 8 waves (wave32).
// Stage 1: build A = P^T (P[m,j] = masked gathered proc_times) in LDS, plus
//          job features f0/f1 and W_node in LDS.
// Stage 2: each wave computes 8 of the 64 (16x16) output tiles of
//          S = A[64x64] @ ma_emb[64x256] with chained v_wmma_f32_16x16x4_f32,
//          then applies the fused node+edge epilogue and writes 3 outputs.
__global__ __launch_bounds__(256) void jssp_dyn_emb_kernel(
    const float* __restrict__ lbs,        // [BS, OO]
    const float* __restrict__ time_v,     // [BS]
    const float* __restrict__ is_ready,   // [BS, OO]
    const float* __restrict__ proc_times, // [BS, MM, OO]
    const float* __restrict__ busy_until, // [BS, MM]
    const float* __restrict__ ma_emb,     // [BS, MM, EE]
    const float* __restrict__ W_node,     // [2, 3*EE]
    const float* __restrict__ W_edge,     // [3]
    const int*   __restrict__ next_op,    // [BS, JJ]
    float* __restrict__ out)              // 3 x [BS, JJ, EE] concatenated
{
    __shared__ float sA[JJ][MM + 1];   // A[j][m] = P^T, padded stride 65
    __shared__ float sW[2][3 * EE];    // W_node
    __shared__ float sF0[JJ];          // lbs feature per job
    __shared__ float sF1[JJ];          // is_ready feature per job
    __shared__ int   sOp[JJ];          // next_op per job

    const int b   = blockIdx.x;
    const int tid = threadIdx.x;
    const float t = time_v[b];

    // ---- Stage 0: per-job gather of node features + op indices ----
    if (tid < JJ) {
        const int op = next_op[b * JJ + tid];
        sOp[tid] = op;
        const float l = lbs[(size_t)b * OO + op] - t;
        sF0[tid] = (l > 0.0f ? l : 0.0f) * INV_SCALE;
        sF1[tid] = is_ready[(size_t)b * OO + op];
    }
    // Stage W_node into LDS (2*768 = 1536 floats, 6 per thread)
    for (int i = tid; i < 2 * 3 * EE; i += 256) {
        sW[i / (3 * EE)][i % (3 * EE)] = W_node[i];
    }
    __syncthreads();

    // ---- Stage 1: build A[j][m] = busy ? 0 : proc_times[b,m,op_j]/SCALE ----
    for (int i = tid; i < JJ * MM; i += 256) {
        const int j = i >> 6;        // / MM
        const int m = i & (MM - 1);  // % MM
        const bool busy = busy_until[b * MM + m] > t;
        float v = 0.0f;
        if (!busy) {
            v = proc_times[((size_t)b * MM + m) * OO + sOp[j]] * INV_SCALE;
        }
        sA[j][m] = v;
    }
    __syncthreads();

    // ---- Stage 2: WMMA tiles ----
    const int wave   = tid >> 5;        // 0..7
    const int lane   = tid & 31;        // 0..31
    const int lane16 = lane & 15;
    const int hi     = (lane >> 4);     // 0 for lanes 0-15, 1 for 16-31

    const float we0 = W_edge[0];
    const float we1 = W_edge[1];
    const float we2 = W_edge[2];

    const float* mb = ma_emb + (size_t)b * MM * EE;
    float* out0 = out;
    float* out1 = out + (size_t)BS * JJ * EE;
    float* out2 = out + (size_t)2 * BS * JJ * EE;

    // 64 tiles: jt in [0,4), et in [0,16); wave w takes tiles w, w+8, ...
    for (int tile = wave; tile < (JJ / 16) * (EE / 16); tile += 8) {
        const int jt = tile >> 4;   // 0..3
        const int et = tile & 15;   // 0..15
        const int jrow = jt * 16 + lane16;   // A row for this lane
        const int ecol = et * 16 + lane16;   // B/D column for this lane

        v8f acc = {};
        #pragma unroll
        for (int k0 = 0; k0 < MM; k0 += 4) {
            const int ka = k0 + (hi << 1);   // lanes 0-15: k0,k0+1 ; 16-31: k0+2,k0+3
            v2f a, bf;
            a.x  = sA[jrow][ka];
            a.y  = sA[jrow][ka + 1];
            bf.x = mb[(size_t)ka * EE + ecol];
            bf.y = mb[(size_t)(ka + 1) * EE + ecol];
            // D = A(16x4) * B(4x16) + C ; exact f32 accumulation
            acc = __builtin_amdgcn_wmma_f32_16x16x4_f32(
                false, a, false, bf, (short)0, acc, false, false);
        }

        // ---- Fused epilogue: node_upd + W_edge[k] * S ----
        #pragma unroll
        for (int r = 0; r < 8; ++r) {
            const int j = jt * 16 + r + (hi << 3);  // M row per C/D layout
            const float s  = acc[r];
            const float f0 = sF0[j];
            const float f1 = sF1[j];
            const int e  = ecol;
            const size_t o = ((size_t)b * JJ + j) * EE + e;
            out0[o] = f0 * sW[0][e]          + f1 * sW[1][e]          + we0 * s;
            out1[o] = f0 * sW[0][EE + e]     + f1 * sW[1][EE + e]     + we1 * s;
            out2[o] = f0 * sW[0][2 * EE + e] + f1 * sW[1][2 * EE + e] + we2 * s;
        }
    }
}

extern "C" void kernel_launch(void* const* d_in, const int* in_sizes, int n_in,
                              void* d_out, int out_size, void* d_ws, size_t ws_size,
                              hipStream_t stream) {
    const float* lbs        = (const float*)d_in[0];
    const float* time_v     = (const float*)d_in[1];
    const float* is_ready   = (const float*)d_in[2];
    const float* proc_times = (const float*)d_in[3];
    const float* busy_until = (const float*)d_in[4];
    const float* ma_emb     = (const float*)d_in[5];
    const float* W_node     = (const float*)d_in[6];
    const float* W_edge     = (const float*)d_in[7];
    const int*   next_op    = (const int*)d_in[8];
    float* out = (float*)d_out;

    jssp_dyn_emb_kernel<<<BS, 256, 0, stream>>>(
        lbs, time_v, is_ready, proc_times, busy_until,
        ma_emb, W_node, W_edge, next_op, out);
}